// CombiIntegral_12283606467928
// MI455X (gfx1250) — compile-verified
//
#include <hip/hip_runtime.h>
#include <stdint.h>
#include <stddef.h>

// CombiIntegral: 9 circular-shift channel blocks (the FFT branch collapses to
// delta-kernel circular convs) + 3 zero-padded 2-tap diff blocks.
// Memory-bound: 32 MiB in, 384 MiB out -> optimize for store bandwidth.

typedef float v4f __attribute__((ext_vector_type(4)));

#define B_  4
#define C_  32
#define H_  256
#define W_  256
#define TM  8            // rows per block tile

__global__ __launch_bounds__(256)
void combi_shift_diff_kernel(const float* __restrict__ x, float* __restrict__ out) {
    const int tid   = threadIdx.x;
    const int colg  = tid & 63;        // 64 column-groups of 4 floats (16 B)
    const int rlane = tid >> 6;        // 0..3
    const int m0    = blockIdx.x * TM; // first output row of tile
    const int bc    = blockIdx.y;      // b*C + ch
    const int b     = bc >> 5;
    const int ch    = bc & (C_ - 1);

    // 10 wrapped input rows: m0-1 .. m0+TM (circular in H)
    __shared__ __align__(16) float smem[(TM + 2) * W_];

    const uint32_t plane_bytes = (uint32_t)(H_ * W_ * 4);
    const uint32_t plane_off   = (uint32_t)bc * plane_bytes;
    // Low 32 bits of a generic pointer to LDS == wave-relative LDS byte offset.
    const uint32_t lds_base    = (uint32_t)(uintptr_t)&smem[0];

    // ---- stage tile: async global -> LDS, 16B per lane, GVS addressing ----
#pragma unroll
    for (int rr = 0; rr < 3; ++rr) {
        const int j = rr * 4 + rlane;          // LDS row 0..9
        if (j < TM + 2) {
            int g = m0 - 1 + j;                // wrapped source row
            g = (g < 0) ? (g + H_) : ((g >= H_) ? (g - H_) : g);
            const uint32_t goff = plane_off + (uint32_t)g * (W_ * 4u) + (uint32_t)colg * 16u;
            const uint32_t la   = lds_base + (uint32_t)j * (W_ * 4u) + (uint32_t)colg * 16u;
            asm volatile("global_load_async_to_lds_b128 %0, %1, %2"
                         :: "v"(la), "v"(goff), "s"(x) : "memory");
        }
    }
    asm volatile("s_wait_asynccnt 0x0" ::: "memory");
    __syncthreads();

    const int n0 = colg * 4;                   // first of 4 output columns
    const int nl = (n0 - 1) & (W_ - 1);        // wrapped left halo column
    const int nr = (n0 + 4) & (W_ - 1);        // wrapped right halo column

    const size_t plane   = (size_t)H_ * W_;
    const size_t bstride = (size_t)C_ * plane;               // one channel-block
    float* const ob = out + ((size_t)b * 12 * C_ + ch) * plane;

#pragma unroll
    for (int rep = 0; rep < 2; ++rep) {
        const int j = rlane + rep * 4;         // tile row 0..7
        const int m = m0 + j;                  // global output row
        const float* rU = &smem[(size_t)j * W_];       // row m-1 (wrapped)
        const float* rC = rU + W_;                     // row m
        const float* rD = rC + W_;                     // row m+1 (wrapped)

        const v4f U  = *(const v4f*)(rU + n0);
        const v4f Cv = *(const v4f*)(rC + n0);
        const v4f D  = *(const v4f*)(rD + n0);
        const float ul = rU[nl], cl = rC[nl], dl = rD[nl];
        const float ur = rU[nr], cr = rC[nr], dr = rD[nr];

        // column-shifted variants (wrapped)
        const v4f LU = {ul, U.x, U.y, U.z};
        const v4f LC = {cl, Cv.x, Cv.y, Cv.z};
        const v4f LD = {dl, D.x, D.y, D.z};
        const v4f RU = {U.y, U.z, U.w, ur};
        const v4f RC = {Cv.y, Cv.z, Cv.w, cr};
        const v4f RD = {D.y, D.z, D.w, dr};

        float* p = ob + (size_t)m * W_ + n0;

        // spectral blocks i=0..8: out_i[m,n] = x[(m+1-r)%H, (n+1-c)%W]
        __builtin_nontemporal_store(RD, (v4f*)(p + 0 * bstride)); // (m+1, n+1)
        __builtin_nontemporal_store(D,  (v4f*)(p + 1 * bstride)); // (m+1, n  )
        __builtin_nontemporal_store(LD, (v4f*)(p + 2 * bstride)); // (m+1, n-1)
        __builtin_nontemporal_store(RC, (v4f*)(p + 3 * bstride)); // (m,   n+1)
        __builtin_nontemporal_store(Cv, (v4f*)(p + 4 * bstride)); // (m,   n  )
        __builtin_nontemporal_store(LC, (v4f*)(p + 5 * bstride)); // (m,   n-1)
        __builtin_nontemporal_store(RU, (v4f*)(p + 6 * bstride)); // (m-1, n+1)
        __builtin_nontemporal_store(U,  (v4f*)(p + 7 * bstride)); // (m-1, n  )
        __builtin_nontemporal_store(LU, (v4f*)(p + 8 * bstride)); // (m-1, n-1)

        // diff blocks (zero padding instead of wrap)
        const v4f zero = (v4f)0.0f;
        v4f v0 = LC;                 if (n0 == 0) v0.x = 0.0f;   // x[m, n-1]
        v4f v1 = (m > 0) ? U  : zero;                            // x[m-1, n]
        v4f v2 = (m > 0) ? LU : zero; if (n0 == 0) v2.x = 0.0f;  // x[m-1, n-1]
        __builtin_nontemporal_store(v0 - Cv, (v4f*)(p +  9 * bstride));
        __builtin_nontemporal_store(v1 - Cv, (v4f*)(p + 10 * bstride));
        __builtin_nontemporal_store(v2 - Cv, (v4f*)(p + 11 * bstride));
    }
}

extern "C" void kernel_launch(void* const* d_in, const int* in_sizes, int n_in,
                              void* d_out, int out_size, void* d_ws, size_t ws_size,
                              hipStream_t stream) {
    (void)in_sizes; (void)n_in; (void)out_size; (void)d_ws; (void)ws_size;
    const float* x = (const float*)d_in[0];
    float* out = (float*)d_out;
    dim3 grid(H_ / TM, B_ * C_);   // 32 row-tiles x 128 (b,ch) planes = 4096 blocks
    combi_shift_diff_kernel<<<grid, 256, 0, stream>>>(x, out);
}